// Model_65335042507143
// MI455X (gfx1250) — compile-verified
//
#include <hip/hip_runtime.h>
#include <hip/hip_bf16.h>

// Ragged gather: out[cu[i]:cu[i+1]] = req_to_token[pool[i], start[i]:start[i]+len[i]]
// Pure data movement (~9.4 MB total) -> bandwidth bound. Use the CDNA5 async
// global<->LDS streaming path (GLOBAL_LOAD_ASYNC_TO_LDS_B32 /
// GLOBAL_STORE_ASYNC_FROM_LDS_B32, ASYNCcnt) so each wave keeps 8x128B
// transfers in flight with zero VGPR staging. 4-byte alignment only (start is
// arbitrary), so B32 is the widest legal async element.

#define ROW_STRIDE 65536   // MAX_CONTEXT from the reference (req_to_token row length)
#define NWAVES 8
#define NBUF   8           // 8 outstanding 128B async ops per wave per phase
#define LANES  32          // wave32 on gfx1250

__device__ __forceinline__ void async_load_dw(unsigned lds_off, unsigned byte_off,
                                              unsigned long long sbase) {
  // global_load_async_to_lds_b32 vdst(lds addr), vaddr(32b offset), saddr(64b base)
  asm volatile("global_load_async_to_lds_b32 %0, %1, %2"
               :: "v"(lds_off), "v"(byte_off), "s"(sbase) : "memory");
}

__device__ __forceinline__ void async_store_dw(unsigned byte_off, unsigned lds_off,
                                               unsigned long long dbase) {
  // global_store_async_from_lds_b32 vaddr(32b offset), vsrc(lds addr), saddr(64b base)
  asm volatile("global_store_async_from_lds_b32 %0, %1, %2"
               :: "v"(byte_off), "v"(lds_off), "s"(dbase) : "memory");
}

__device__ __forceinline__ void wait_async0() {
  asm volatile("s_wait_asynccnt 0x0" ::: "memory");
}

__global__ void __launch_bounds__(NWAVES * LANES)
ragged_gather_async(const int* __restrict__ req_to_token,
                    const int* __restrict__ req_pool_indices,
                    const int* __restrict__ chunk_starts,
                    const int* __restrict__ chunk_cu,
                    int* __restrict__ out,
                    int row_stride) {
  __shared__ int tile[NWAVES * NBUF * LANES];   // 8 KB: private 1 KB region per wave

  const int b    = blockIdx.x;
  const int wave = threadIdx.x >> 5;
  const int lane = threadIdx.x & 31;

  // Uniform per-block scalars (compiler emits s_load for these).
  const int row   = req_pool_indices[b];
  const int start = chunk_starts[b];
  const int c0    = chunk_cu[b];
  const int c1    = chunk_cu[b + 1];
  const int len   = c1 - c0;
  if (len <= 0) return;

  const int* src = req_to_token + (size_t)row * (size_t)row_stride + (size_t)start;
  int*       dst = out + c0;

  // Contiguous per-wave span of this segment (uniform within a wave).
  const int span = (len + NWAVES - 1) / NWAVES;
  const int s0   = wave * span;
  int s1         = s0 + span;
  if (s1 > len) s1 = len;
  if (s0 >= s1) return;   // no barriers used anywhere -> early exit is safe

  // Low 32 bits of a flat LDS pointer == wave-relative LDS byte offset.
  const unsigned lds_wave =
      (unsigned)(size_t)tile + (unsigned)(wave * NBUF * LANES * 4) + (unsigned)(lane * 4);
  const unsigned long long sb = (unsigned long long)(size_t)src;
  const unsigned long long db = (unsigned long long)(size_t)dst;

  int p = s0;
  for (; p + NBUF * LANES <= s1; p += NBUF * LANES) {
#pragma unroll
    for (int k = 0; k < NBUF; ++k)
      async_load_dw(lds_wave + (unsigned)(k * LANES * 4),
                    (unsigned)((p + k * LANES + lane) * 4), sb);
    wait_async0();                       // all 8 loads landed in LDS
#pragma unroll
    for (int k = 0; k < NBUF; ++k)
      async_store_dw((unsigned)((p + k * LANES + lane) * 4),
                     lds_wave + (unsigned)(k * LANES * 4), db);
    wait_async0();                       // stores drained -> LDS buffers reusable
  }

  // Tail (< NBUF*LANES dwords): plain coalesced dword copy.
  for (int i = p + lane; i < s1; i += LANES)
    dst[i] = src[i];
}

extern "C" void kernel_launch(void* const* d_in, const int* in_sizes, int n_in,
                              void* d_out, int out_size, void* d_ws, size_t ws_size,
                              hipStream_t stream) {
  const int* req_to_token     = (const int*)d_in[0];
  const int* req_pool_indices = (const int*)d_in[1];
  const int* chunk_starts     = (const int*)d_in[2];
  // d_in[3] = chunk_seq_lens (unused: lengths are derived from the cu prefix sums)
  const int* chunk_cu         = (const int*)d_in[4];
  int*       out              = (int*)d_out;   // int32 output, bit-compatible reinterpret

  const int batch = in_sizes[1];               // BATCH = len(req_pool_indices)
  (void)n_in; (void)out_size; (void)d_ws; (void)ws_size;

  ragged_gather_async<<<batch, NWAVES * LANES, 0, stream>>>(
      req_to_token, req_pool_indices, chunk_starts, chunk_cu, out, ROW_STRIDE);
}